// Basic_Aggregator_40922448396317
// MI455X (gfx1250) — compile-verified
//
#include <hip/hip_runtime.h>
#include <hip/hip_bf16.h>

// Problem shape (fixed by the reference setup_inputs()):
//   node_j : [16, 20000, 64] fp32
//   out    : [16, 20000, 64] fp32  (per-batch column sum broadcast to all rows)
#define BATCHES 16
#define NROWS   20000
#define DIM     64
#define CHUNKS  125                      // 16*125*4 = 8000 waves in phase 1
#define ROWS_PER_CHUNK (NROWS / CHUNKS)  // 160, divisible by 4 (WMMA K)

typedef __attribute__((ext_vector_type(2))) float v2f;
typedef __attribute__((ext_vector_type(8))) float v8f;

// ---------------------------------------------------------------------------
// Phase 1: per-(batch, chunk) partial column sums via V_WMMA_F32_16X16X4_F32.
// A = 16x4 ones, B = 4x16 tile of node rows  =>  every row of D is the column
// sum of the tile; C accumulates across tiles. One wave owns one 16-column
// slice; block = 128 threads = 4 waves = all 64 columns.
// ---------------------------------------------------------------------------
__global__ void __launch_bounds__(128)
colsum_wmma_kernel(const float* __restrict__ node, float* __restrict__ partial) {
    const int b     = blockIdx.y;
    const int chunk = blockIdx.x;
    const int wave  = threadIdx.x >> 5;        // 0..3 -> column slice
    const int lane  = threadIdx.x & 31;
    const int col   = (wave << 4) + (lane & 15);
    // 32-bit B layout: lanes 0-15 carry K={0,1}, lanes 16-31 carry K={2,3}
    const int krow  = (lane < 16) ? 0 : 2;

    const float* base = node + (size_t)b * NROWS * DIM + col;

    v2f a; a.x = 1.0f; a.y = 1.0f;             // all-ones A matrix
    v8f c = {};                                 // accumulator (16x16 f32)

    const int r0 = chunk * ROWS_PER_CHUNK;
    #pragma unroll 4
    for (int r = r0; r < r0 + ROWS_PER_CHUNK; r += 4) {
        const float* p = base + (size_t)(r + krow) * DIM;
        v2f bm;
        bm.x = p[0];      // row r+krow,   this lane's column
        bm.y = p[DIM];    // row r+krow+1, this lane's column
        c = __builtin_amdgcn_wmma_f32_16x16x4_f32(
                /*neg_a=*/false, a, /*neg_b=*/false, bm,
                /*c_mod=*/(short)0, c, /*reuse_a=*/false, /*reuse_b=*/false);
    }

    // D row M=0 lives in c[0], lanes 0..15 (N = lane). All D rows are equal
    // (A is ones), so one row suffices.
    if (lane < 16) {
        partial[((size_t)b * CHUNKS + chunk) * DIM + col] = c[0];
    }
}

// ---------------------------------------------------------------------------
// Phase 2: fold CHUNKS partials per (batch, col) in a fixed order
// (deterministic; no atomics). 16 blocks x 64 threads.
// ---------------------------------------------------------------------------
__global__ void __launch_bounds__(64)
reduce_partials_kernel(const float* __restrict__ partial, float* __restrict__ sums) {
    const int b = blockIdx.x;
    const int d = threadIdx.x;                 // 0..63
    float acc = 0.0f;
    #pragma unroll 5
    for (int ch = 0; ch < CHUNKS; ++ch)
        acc += partial[((size_t)b * CHUNKS + ch) * DIM + d];
    sums[b * DIM + d] = acc;
}

// ---------------------------------------------------------------------------
// Phase 3: broadcast sums back to every row. Pure float4 streaming stores;
// the 4 KB of sums stays L2/WGP$-resident, so stores run at HBM rate.
// grid = (NROWS*16/256, BATCHES), 256 threads.
// ---------------------------------------------------------------------------
__global__ void __launch_bounds__(256)
broadcast_kernel(const float* __restrict__ sums, float4* __restrict__ out) {
    const int    b   = blockIdx.y;
    const size_t idx = (size_t)blockIdx.x * blockDim.x + threadIdx.x; // 0..NROWS*16-1
    const int    q   = (int)(idx & 15);        // which float4 of the 64-vector
    const float4* s4 = (const float4*)(sums + b * DIM);
    out[(size_t)b * (NROWS * (DIM / 4)) + idx] = s4[q];
}

// ---------------------------------------------------------------------------
extern "C" void kernel_launch(void* const* d_in, const int* in_sizes, int n_in,
                              void* d_out, int out_size, void* d_ws, size_t ws_size,
                              hipStream_t stream) {
    const float* node = (const float*)d_in[0];
    (void)d_in; (void)in_sizes; (void)n_in; (void)out_size; (void)ws_size;

    // Workspace layout (floats):
    //   [0, BATCHES*CHUNKS*DIM)                       : phase-1 partials (512 KB)
    //   [BATCHES*CHUNKS*DIM, +BATCHES*DIM)            : final sums (4 KB)
    float* partial = (float*)d_ws;
    float* sums    = partial + (size_t)BATCHES * CHUNKS * DIM;

    {   // Phase 1: WMMA ones-matmul partial reduction
        dim3 grid(CHUNKS, BATCHES);
        colsum_wmma_kernel<<<grid, 128, 0, stream>>>(node, partial);
    }
    {   // Phase 2: deterministic combine
        reduce_partials_kernel<<<BATCHES, 64, 0, stream>>>(partial, sums);
    }
    {   // Phase 3: broadcast (write-bound streaming phase)
        dim3 grid((NROWS * (DIM / 4) * 4) / 256 / 4 * 4 / 4, BATCHES);
        // NROWS*16 float4 per batch = 320000; 320000/256 = 1250 blocks exactly.
        grid.x = (NROWS * 16) / 256;
        broadcast_kernel<<<grid, 256, 0, stream>>>(sums, (float4*)d_out);
    }
}